// PointCrop2D_27943057227941
// MI455X (gfx1250) — compile-verified
//
#include <hip/hip_runtime.h>
#include <cstdint>

// PointCrop2D: per-batch dynamic 224x224x3 crop with fill = (0-0.45)/0.225 = -2.0
// Pure data movement (~38.6 MB HBM traffic -> ~1.7us floor at 23.3 TB/s).
// Uses CDNA5 async global<->LDS path: global_load_async_to_lds_b32 /
// global_store_async_from_lds_b32 + s_wait_asynccnt / s_wait_dscnt.

#define CROP   224
#define DIA    112
#define ROW_F  (CROP * 3)   // 672 dwords per output row (contiguous)
#define IMG_W  512
#define IMG_H  512
#define FILLV  (-2.0f)      // (0.0 - MEAN) / STD

__global__ __launch_bounds__(CROP)
void pointcrop2d_kernel(const float* __restrict__ points,
                        const float* __restrict__ images,
                        float* __restrict__ out)
{
    __shared__ float srow[ROW_F];     // 2688 B row staging buffer

    const int i = blockIdx.x;         // output row 0..223
    const int b = blockIdx.y;         // batch
    const int t = threadIdx.x;        // 0..223

    // Per-batch crop center: points[b,0,0] -> x, points[b,1,0] -> y (uniform).
    int x = (int)points[b * 9 + 0];
    int y = (int)points[b * 9 + 3];
    x = min(max(x, 1), IMG_W - 2);
    y = min(max(y, 1), IMG_H - 2);

    const int  sr  = y - DIA + i;                       // source row
    const bool rok = (sr >= 0) & (sr <= IMG_H - 2);     // row 511 / OOB -> fill
    const int  srcr = min(max(sr, 0), IMG_H - 2);       // clamped (address-safe)

    const float* imgRow = images + ((size_t)b * IMG_H + srcr) * (size_t)(IMG_W * 3);
    float*       outRow = out    + ((size_t)b * CROP  + i   ) * (size_t)ROW_F;

    // LDS byte offset of the staging buffer: generic LDS pointers carry the
    // 32-bit LDS offset in their low dword (flat->LDS uses addr[31:0]).
    const unsigned ldsBase = (unsigned)(uintptr_t)&srow[0];

    unsigned     ldsOff[3];
    const float* gsrc[3];
    float*       gdst[3];
    bool         ok[3];

#pragma unroll
    for (int k = 0; k < 3; ++k) {
        const int d   = t + k * CROP;     // dword index in row, 0..671
        const int p   = d / 3;            // pixel 0..223
        const int c   = d - 3 * p;        // channel 0..2
        const int sp  = x - DIA + p;      // source col of this pixel
        const bool cok = (sp >= 0) & (sp <= IMG_W - 2);
        const int spc = min(max(sp, 0), IMG_W - 2);
        ok[k]     = rok & cok;
        gsrc[k]   = imgRow + spc * 3 + c; // always a legal address
        gdst[k]   = outRow + d;
        ldsOff[k] = ldsBase + (unsigned)(d * 4);
    }

    // 1) Async gather: global -> LDS (clamped addresses; OOB fixed up below).
#pragma unroll
    for (int k = 0; k < 3; ++k) {
        asm volatile("global_load_async_to_lds_b32 %0, %1, off"
                     :
                     : "v"(ldsOff[k]), "v"(gsrc[k])
                     : "memory");
    }
    asm volatile("s_wait_asynccnt 0" ::: "memory");

    // 2) Overwrite this lane's OOB dwords with the fill value (ds_store_b32).
#pragma unroll
    for (int k = 0; k < 3; ++k) {
        if (!ok[k]) srow[t + k * CROP] = FILLV;
    }
    asm volatile("s_wait_dscnt 0" ::: "memory");

    // 3) Async scatter: LDS -> contiguous output row (coalesced).
    //    Each lane stores exactly the dwords it produced -> no barrier needed.
#pragma unroll
    for (int k = 0; k < 3; ++k) {
        asm volatile("global_store_async_from_lds_b32 %0, %1, off"
                     :
                     : "v"(gdst[k]), "v"(ldsOff[k])
                     : "memory");
    }
    // S_ENDPGM performs an implicit wait-idle; outstanding ASYNCcnt drains.
}

extern "C" void kernel_launch(void* const* d_in, const int* in_sizes, int n_in,
                              void* d_out, int out_size, void* d_ws, size_t ws_size,
                              hipStream_t stream)
{
    const float* points = (const float*)d_in[0];   // (B, 3, 3) f32
    const float* images = (const float*)d_in[1];   // (B, 512, 512, 3) f32
    float*       out    = (float*)d_out;           // (B, 224, 224, 3) f32

    const int B = in_sizes[0] / 9;                 // 32

    dim3 grid(CROP, B);   // one block per (output row, batch)
    dim3 block(CROP);     // 224 threads = 7 wave32
    pointcrop2d_kernel<<<grid, block, 0, stream>>>(points, images, out);
}